// TopKSAE_21431886807365
// MI455X (gfx1250) — compile-verified
//
#include <hip/hip_runtime.h>

typedef __attribute__((ext_vector_type(2))) float v2f;
typedef __attribute__((ext_vector_type(4))) float v4f;
typedef __attribute__((ext_vector_type(8))) float v8f;

#define B_ROWS 8192
#define D_IN   768
#define N_LAT  32768
#define TOPK   32

#define BM 64
#define BN 128
#define KC 32
#define KS (KC + 4)  // LDS row stride 36: k-contiguous frags, 16B-aligned v4f
                     // stores, conflict-free b64 frag reads

// ---------------------------------------------------------------------------
// Kernel 1: z = relu((x - bias) @ W_enc^T), fp32 WMMA 16x16x4.
// Grid: (N_LAT/BN, B_ROWS/BM), 256 threads = 8 waves tiled 2(M) x 4(N).
// Each wave computes a 32x32 tile = 4 accumulators.
// Pipeline: global->reg prefetch of chunk k+1 overlaps the 32-WMMA compute
// of chunk k; reg->LDS stores happen between barriers.
// ---------------------------------------------------------------------------
__global__ __launch_bounds__(256) void sae_enc_gemm(
    const float* __restrict__ x,     // [B_ROWS, D_IN]
    const float* __restrict__ W,     // W_enc [N_LAT, D_IN]
    const float* __restrict__ bias,  // [D_IN]
    float* __restrict__ z)           // [B_ROWS, N_LAT]
{
  __shared__ float sA[BM][KS];   // (x - bias) tile, [m][k]
  __shared__ float sB[BN][KS];   // W_enc tile, [n][k]  (k contiguous)

  const int tid  = threadIdx.x;
  const int lane = tid & 31;
  const int wid  = tid >> 5;            // 0..7
  const int wm   = (wid >> 2) * 32;     // 0 or 32
  const int wn   = (wid & 3) * 32;      // 0,32,64,96
  const int lm   = lane & 15;
  const int half = lane >> 4;           // 0: lanes 0-15, 1: lanes 16-31

  const int gm0 = blockIdx.y * BM;
  const int gn0 = blockIdx.x * BN;

  v8f c00 = {};  // rows wm..wm+15,  cols wn..wn+15
  v8f c01 = {};  // rows wm..wm+15,  cols wn+16..wn+31
  v8f c10 = {};  // rows wm+16..+31, cols wn..wn+15
  v8f c11 = {};  // rows wm+16..+31, cols wn+16..wn+31

  const int ar = tid >> 3;        // 0..31
  const int ac = (tid & 7) * 4;   // 0,4,...,28
  const int bn = tid >> 1;        // 0..127
  const int bk = (tid & 1) * 16;  // 0 or 16

  const float* xa0 = x + (size_t)(gm0 + ar) * D_IN + ac;
  const float* xa1 = x + (size_t)(gm0 + ar + 32) * D_IN + ac;
  const float* wb  = W + (size_t)(gn0 + bn) * D_IN + bk;

  // Register prefetch buffers (next K-chunk).
  v4f ra0, ra1, rb0, rb1, rb2, rb3, rbv;
  {
    rbv = *(const v4f*)(bias + ac);
    ra0 = *(const v4f*)(xa0);
    ra1 = *(const v4f*)(xa1);
    rb0 = *(const v4f*)(wb + 0);
    rb1 = *(const v4f*)(wb + 4);
    rb2 = *(const v4f*)(wb + 8);
    rb3 = *(const v4f*)(wb + 12);
  }

  for (int k0 = 0; k0 < D_IN; k0 += KC) {
    __syncthreads();  // previous chunk's LDS reads complete (no-op on iter 0)

    *(v4f*)&sA[ar][ac]        = ra0 - rbv;
    *(v4f*)&sA[ar + 32][ac]   = ra1 - rbv;
    *(v4f*)&sB[bn][bk + 0]    = rb0;
    *(v4f*)&sB[bn][bk + 4]    = rb1;
    *(v4f*)&sB[bn][bk + 8]    = rb2;
    *(v4f*)&sB[bn][bk + 12]   = rb3;

    __syncthreads();  // tile visible

    // Issue next chunk's global loads; they retire during the compute below.
    const int kn = k0 + KC;
    if (kn < D_IN) {
      rbv = *(const v4f*)(bias + kn + ac);
      ra0 = *(const v4f*)(xa0 + kn);
      ra1 = *(const v4f*)(xa1 + kn);
      rb0 = *(const v4f*)(wb + kn + 0);
      rb1 = *(const v4f*)(wb + kn + 4);
      rb2 = *(const v4f*)(wb + kn + 8);
      rb3 = *(const v4f*)(wb + kn + 12);
    }

    #pragma unroll
    for (int kk = 0; kk < KC; kk += 4) {
      // 16x4 A / 4x16 B frags: lanes 0-15 hold K = kk,kk+1; lanes 16-31 K+2.
      const int krow = kk + 2 * half;
      v2f a0 = *(const v2f*)&sA[wm + lm][krow];
      v2f a1 = *(const v2f*)&sA[wm + 16 + lm][krow];
      v2f b0 = *(const v2f*)&sB[wn + lm][krow];
      v2f b1 = *(const v2f*)&sB[wn + 16 + lm][krow];
      c00 = __builtin_amdgcn_wmma_f32_16x16x4_f32(false, a0, false, b0,
                                                  (short)0, c00, false, false);
      c10 = __builtin_amdgcn_wmma_f32_16x16x4_f32(false, a1, false, b0,
                                                  (short)0, c10, false, false);
      c01 = __builtin_amdgcn_wmma_f32_16x16x4_f32(false, a0, false, b1,
                                                  (short)0, c01, false, false);
      c11 = __builtin_amdgcn_wmma_f32_16x16x4_f32(false, a1, false, b1,
                                                  (short)0, c11, false, false);
    }
  }

  // Epilogue: ReLU + store. C layout: VGPR r -> M = half*8 + r, N = lane&15.
  const int col0 = gn0 + wn + lm;
  #pragma unroll
  for (int r = 0; r < 8; ++r) {
    const int row0 = gm0 + wm + half * 8 + r;
    z[(size_t)row0 * N_LAT + col0]             = fmaxf(c00[r], 0.0f);
    z[(size_t)row0 * N_LAT + col0 + 16]        = fmaxf(c01[r], 0.0f);
    z[(size_t)(row0 + 16) * N_LAT + col0]      = fmaxf(c10[r], 0.0f);
    z[(size_t)(row0 + 16) * N_LAT + col0 + 16] = fmaxf(c11[r], 0.0f);
  }
}

// ---------------------------------------------------------------------------
// Kernel 2: exact per-row top-32 via 8-bit radix select on float bits
// (values >= 0 post-ReLU => bits monotone in value). Masks z in place and
// emits a deterministic, index-ordered (val, idx) list of <=32 entries.
// One block (256 threads) per row.
// ---------------------------------------------------------------------------
__global__ __launch_bounds__(256) void sae_topk(
    float* __restrict__ z,
    float* __restrict__ lv, int* __restrict__ li, int* __restrict__ lc)
{
  const int row = blockIdx.x;
  const int tid = threadIdx.x;
  float* zr = z + (size_t)row * N_LAT;

  __shared__ unsigned hist[256];
  __shared__ unsigned sh_prefix;
  __shared__ int sh_k;

  unsigned prefix = 0;
  int k = TOPK;
  for (int pass = 0; pass < 4; ++pass) {
    const int shift = 24 - 8 * pass;
    const unsigned pmask = pass ? (0xFFFFFFFFu << (shift + 8)) : 0u;
    hist[tid] = 0;
    __syncthreads();
    for (int i = tid; i < N_LAT; i += 256) {
      const unsigned b = __float_as_uint(zr[i]);
      if ((b & pmask) == prefix)
        atomicAdd(&hist[(b >> shift) & 255u], 1u);
    }
    __syncthreads();
    if (tid == 0) {
      unsigned cum = 0;
      int d;
      for (d = 255; d > 0; --d) {
        if (cum + hist[d] >= (unsigned)k) break;
        cum += hist[d];
      }
      sh_prefix = prefix | ((unsigned)d << shift);
      sh_k = k - (int)cum;
    }
    __syncthreads();
    prefix = sh_prefix;
    k = sh_k;
    __syncthreads();  // hist reused next pass
  }
  const unsigned T = prefix;  // bit pattern of the 32nd-largest element
  const int kf = k;           // # of ties (== T) to admit, lowest index first

  // Deterministic selection: contiguous 128-elem chunks + LDS prefix scans.
  __shared__ int sgt[256], stie[256];
  const int CH = N_LAT / 256;
  const int base = tid * CH;
  int loc_gt = 0, loc_tie = 0;
  for (int i = 0; i < CH; ++i) {
    const unsigned b = __float_as_uint(zr[base + i]);
    loc_gt  += (b > T);
    loc_tie += (b == T);
  }
  sgt[tid] = loc_gt;
  stie[tid] = loc_tie;
  __syncthreads();
  for (int off = 1; off < 256; off <<= 1) {
    int a = 0, t = 0;
    if (tid >= off) { a = sgt[tid - off]; t = stie[tid - off]; }
    __syncthreads();
    sgt[tid] += a;
    stie[tid] += t;
    __syncthreads();
  }
  int gt_run  = sgt[tid]  - loc_gt;   // exclusive prefix (by index)
  int tie_run = stie[tid] - loc_tie;
  const int gt_total  = sgt[255];
  const int tie_total = stie[255];

  for (int i = 0; i < CH; ++i) {
    const int idx = base + i;
    const float v = zr[idx];
    const unsigned b = __float_as_uint(v);
    if (b > T) {
      const int slot = gt_run + min(tie_run, kf);
      lv[(size_t)row * TOPK + slot] = v;
      li[(size_t)row * TOPK + slot] = idx;
      ++gt_run;
    } else if (b == T) {
      if (tie_run < kf) {
        const int slot = gt_run + tie_run;
        lv[(size_t)row * TOPK + slot] = v;
        li[(size_t)row * TOPK + slot] = idx;
      } else {
        zr[idx] = 0.0f;
      }
      ++tie_run;
    } else {
      zr[idx] = 0.0f;
    }
  }
  if (tid == 0) lc[row] = gt_total + min(tie_total, kf);
}

// ---------------------------------------------------------------------------
// Kernel 3: sparse decode. W_dec[:, j] == W_enc[j, :] (contiguous rows), so
// recon[b, :] = bias + sum_k val_k * W_enc[idx_k, :].  One block per row.
// ---------------------------------------------------------------------------
__global__ __launch_bounds__(256) void sae_decode(
    const float* __restrict__ W,     // W_enc [N_LAT, D_IN]
    const float* __restrict__ bias,
    const float* __restrict__ lv, const int* __restrict__ li,
    const int* __restrict__ lc,
    float* __restrict__ recon)       // [B_ROWS, D_IN]
{
  const int row = blockIdx.x;
  const int tid = threadIdx.x;
  __shared__ float sv[TOPK];
  __shared__ int   si[TOPK];
  if (tid < TOPK) {
    const int c = lc[row];
    sv[tid] = (tid < c) ? lv[(size_t)row * TOPK + tid] : 0.0f;
    si[tid] = (tid < c) ? li[(size_t)row * TOPK + tid] : 0;
  }
  __syncthreads();
  for (int d = tid; d < D_IN; d += 256) {
    float acc = bias[d];
    #pragma unroll
    for (int j = 0; j < TOPK; ++j)
      acc += sv[j] * W[(size_t)si[j] * D_IN + d];
    recon[(size_t)row * D_IN + d] = acc;
  }
}

extern "C" void kernel_launch(void* const* d_in, const int* in_sizes, int n_in,
                              void* d_out, int out_size, void* d_ws, size_t ws_size,
                              hipStream_t stream) {
  (void)in_sizes; (void)n_in; (void)out_size; (void)ws_size;
  const float* x     = (const float*)d_in[0];
  const float* W_enc = (const float*)d_in[1];
  // d_in[2] = W_dec is not needed: W_dec^T rows == W_enc rows.
  const float* bias  = (const float*)d_in[3];

  float* out   = (float*)d_out;
  float* out_x = out;                                   // [8192, 768]
  float* out_z = out + (size_t)B_ROWS * D_IN;           // [8192, 32768]
  float* out_r = out_z + (size_t)B_ROWS * N_LAT;        // [8192, 768]

  float* lv = (float*)d_ws;                             // [8192, 32] values
  int*   li = (int*)(lv + (size_t)B_ROWS * TOPK);       // [8192, 32] indices
  int*   lc = li + (size_t)B_ROWS * TOPK;               // [8192] counts

  hipMemcpyAsync(out_x, x, (size_t)B_ROWS * D_IN * sizeof(float),
                 hipMemcpyDeviceToDevice, stream);

  sae_enc_gemm<<<dim3(N_LAT / BN, B_ROWS / BM), 256, 0, stream>>>(
      x, W_enc, bias, out_z);
  sae_topk<<<B_ROWS, 256, 0, stream>>>(out_z, lv, li, lc);
  sae_decode<<<B_ROWS, 256, 0, stream>>>(W_enc, bias, lv, li, lc, out_r);
}